// CharRNN_86792699118061
// MI455X (gfx1250) — compile-verified
//
#include <hip/hip_runtime.h>
#include <hip/hip_bf16.h>
#include <math.h>

typedef __attribute__((ext_vector_type(2))) float v2f;
typedef __attribute__((ext_vector_type(8))) float v8f;

#define RNN_B 128
#define RNN_L 1024
#define RNN_V 32
#define RNN_E 256
#define RNN_H 1024

// ---------- 1x1 tile helper (unpacked B), used only for the tiny one-shot
// embedding-table GEMM ----------
__device__ __forceinline__ v8f tile_gemm_f32(const float* __restrict__ A, int lda,
                                             const float* __restrict__ Bm, int ldb,
                                             int mbase, int nbase, int K) {
  const int lane = threadIdx.x & 31;
  const int half = lane >> 4;
  const int l15  = lane & 15;
  const float* ap = A + (size_t)(mbase + l15) * lda + 2 * half;
  const float* bp = Bm + (size_t)(2 * half) * ldb + (nbase + l15);
  v8f acc = {0.f, 0.f, 0.f, 0.f, 0.f, 0.f, 0.f, 0.f};
#pragma unroll 8
  for (int k = 0; k < K; k += 4) {
    v2f a, b;
    a.x = ap[k];
    a.y = ap[k + 1];
    b.x = bp[(size_t)k * ldb];
    b.y = bp[(size_t)(k + 1) * ldb];
    acc = __builtin_amdgcn_wmma_f32_16x16x4_f32(false, a, false, b, (short)0, acc,
                                                false, false);
  }
  return acc;
}

// ---------- 2x2 register-blocked tile GEMM with K-pair-packed B ----------
// Bp[k2*ldb + n] = {B[2*k2][n], B[2*k2+1][n]}  (ldb = N, in v2f elements)
// Per k-step (K += 4): 4 b64 loads feed 4 WMMAs (1 vmem op per WMMA).
__device__ __forceinline__ void tile_gemm_2x2(const float* __restrict__ A, int lda,
                                              const v2f* __restrict__ Bp, int ldb,
                                              int mbase, int nbase, int K,
                                              v8f acc[4]) {
  const int lane = threadIdx.x & 31;
  const int half = lane >> 4;
  const int l15  = lane & 15;
  const float* a0p = A + (size_t)(mbase + l15) * lda + 2 * half;
  const float* a1p = a0p + (size_t)16 * lda;
  const v2f* b0p = Bp + (size_t)half * ldb + (nbase + l15);
  const v2f* b1p = b0p + 16;
#pragma unroll 4
  for (int k2 = 0; k2 < K / 2; k2 += 2) {  // k2 = K-pair index, covers K += 4
    const v2f a0 = *(const v2f*)(a0p + 2 * k2);
    const v2f a1 = *(const v2f*)(a1p + 2 * k2);
    const v2f b0 = b0p[(size_t)k2 * ldb];
    const v2f b1 = b1p[(size_t)k2 * ldb];
    acc[0] = __builtin_amdgcn_wmma_f32_16x16x4_f32(false, a0, false, b0, (short)0,
                                                   acc[0], false, false);
    acc[1] = __builtin_amdgcn_wmma_f32_16x16x4_f32(false, a0, false, b1, (short)0,
                                                   acc[1], false, false);
    acc[2] = __builtin_amdgcn_wmma_f32_16x16x4_f32(false, a1, false, b0, (short)0,
                                                   acc[2], false, false);
    acc[3] = __builtin_amdgcn_wmma_f32_16x16x4_f32(false, a1, false, b1, (short)0,
                                                   acc[3], false, false);
  }
}

// Pack W[K][N] (row-major) into K-pair v2f layout: Wp[k2][n] = {W[2k2][n], W[2k2+1][n]}
__global__ void __launch_bounds__(256)
pack_pairs_kernel(const float* __restrict__ W, v2f* __restrict__ Wp, int K, int N) {
  const int idx = blockIdx.x * 256 + threadIdx.x;
  const int tot = (K / 2) * N;
  if (idx < tot) {
    const int k2 = idx / N;
    const int n  = idx - k2 * N;
    v2f p;
    p.x = W[(size_t)(2 * k2) * N + n];
    p.y = W[(size_t)(2 * k2 + 1) * N + n];
    Wp[idx] = p;
  }
}

// table[v][h] = (embedding @ W_e)[v][h] + b_h[h]   (V=32 rows, K=E=256)
__global__ void __launch_bounds__(256)
embed_table_kernel(const float* __restrict__ emb, const float* __restrict__ We,
                   const float* __restrict__ bh, float* __restrict__ table) {
  const int wave = threadIdx.x >> 5;
  const int lane = threadIdx.x & 31;
  const int half = lane >> 4;
  const int l15  = lane & 15;
  const int tile = blockIdx.x * 8 + wave;   // 0..127
  const int mt = tile >> 6;                 // 0..1  (V tiles)
  const int nt = tile & 63;                 // 0..63 (H tiles)
  v8f acc = tile_gemm_f32(emb, RNN_E, We, RNN_H, mt * 16, nt * 16, RNN_E);
  const int n  = nt * 16 + l15;
  const int v0 = mt * 16 + half * 8;
#pragma unroll
  for (int r = 0; r < 8; ++r)
    table[(size_t)(v0 + r) * RNN_H + n] = acc[r] + bh[n];
}

// Step t: blocks 0..15 : h_out = tanh(table[x[:,t]] + h_in @ W_h)   (2x2 blocked)
//         block  16    : logits[:, t-1, :] = h_in @ W_o + b_o       (if t > 0)
__global__ void __launch_bounds__(256)
rnn_step_kernel(const float* __restrict__ h_in, float* __restrict__ h_out,
                const float* __restrict__ table, const int* __restrict__ x,
                const v2f* __restrict__ Whp, const v2f* __restrict__ Wop,
                const float* __restrict__ bo, float* __restrict__ logits, int t) {
  const int wave = threadIdx.x >> 5;
  const int lane = threadIdx.x & 31;
  const int half = lane >> 4;
  const int l15  = lane & 15;
  v8f acc[4];
#pragma unroll
  for (int i = 0; i < 4; ++i) acc[i] = (v8f){0.f,0.f,0.f,0.f,0.f,0.f,0.f,0.f};

  if (blockIdx.x < 16) {
    const int task  = blockIdx.x * 8 + wave;  // 0..127
    const int mtb   = task >> 5;              // 0..3  (32-row B-blocks)
    const int ntb   = task & 31;              // 0..31 (32-col H-blocks)
    const int mbase = mtb * 32;
    const int nbase = ntb * 32;
    tile_gemm_2x2(h_in, RNN_H, Whp, RNN_H, mbase, nbase, RNN_H, acc);
    const int n0  = nbase + l15;
    const int b00 = mbase + half * 8;
#pragma unroll
    for (int mi = 0; mi < 2; ++mi) {
#pragma unroll
      for (int r = 0; r < 8; ++r) {
        const int b    = b00 + mi * 16 + r;
        const int xidx = x[(size_t)b * RNN_L + t];
#pragma unroll
        for (int ni = 0; ni < 2; ++ni) {
          const int n = n0 + ni * 16;
          const float xin = table[(size_t)xidx * RNN_H + n];
          h_out[(size_t)b * RNN_H + n] = tanhf(acc[mi * 2 + ni][r] + xin);
        }
      }
    }
  } else if (t > 0 && wave < 4) {
    // logits for step t-1: M = 128 (4 waves x 32 rows), N = V = 32 (one 2x2 task)
    const int mbase = wave * 32;
    tile_gemm_2x2(h_in, RNN_H, Wop, RNN_V, mbase, 0, RNN_H, acc);
    const int n0  = l15;
    const int b00 = mbase + half * 8;
#pragma unroll
    for (int mi = 0; mi < 2; ++mi)
#pragma unroll
      for (int r = 0; r < 8; ++r) {
        const int b = b00 + mi * 16 + r;
#pragma unroll
        for (int ni = 0; ni < 2; ++ni) {
          const int n = n0 + ni * 16;
          logits[(size_t)b * RNN_L * RNN_V + (size_t)(t - 1) * RNN_V + n] =
              acc[mi * 2 + ni][r] + bo[n];
        }
      }
  }
}

// logits[:, L-1, :] from the final hidden state (1 block, 4 active waves).
__global__ void __launch_bounds__(256)
final_logits_kernel(const float* __restrict__ h_fin, const v2f* __restrict__ Wop,
                    const float* __restrict__ bo, float* __restrict__ logits) {
  const int wave = threadIdx.x >> 5;
  if (wave >= 4) return;
  const int lane = threadIdx.x & 31;
  const int half = lane >> 4;
  const int l15  = lane & 15;
  v8f acc[4];
#pragma unroll
  for (int i = 0; i < 4; ++i) acc[i] = (v8f){0.f,0.f,0.f,0.f,0.f,0.f,0.f,0.f};
  const int mbase = wave * 32;
  tile_gemm_2x2(h_fin, RNN_H, Wop, RNN_V, mbase, 0, RNN_H, acc);
  const int n0  = l15;
  const int b00 = mbase + half * 8;
#pragma unroll
  for (int mi = 0; mi < 2; ++mi)
#pragma unroll
    for (int r = 0; r < 8; ++r) {
      const int b = b00 + mi * 16 + r;
#pragma unroll
      for (int ni = 0; ni < 2; ++ni) {
        const int n = n0 + ni * 16;
        logits[(size_t)b * RNN_L * RNN_V + (size_t)(RNN_L - 1) * RNN_V + n] =
            acc[mi * 2 + ni][r] + bo[n];
      }
    }
}

extern "C" void kernel_launch(void* const* d_in, const int* in_sizes, int n_in,
                              void* d_out, int out_size, void* d_ws, size_t ws_size,
                              hipStream_t stream) {
  (void)in_sizes; (void)n_in; (void)out_size; (void)ws_size;
  const int*   x      = (const int*)d_in[0];     // [B, L] int32
  const float* hidden = (const float*)d_in[1];   // [B, H]
  const float* emb    = (const float*)d_in[2];   // [V, E]
  const float* We     = (const float*)d_in[3];   // [E, H]
  const float* Wh     = (const float*)d_in[4];   // [H, H]
  const float* bh     = (const float*)d_in[5];   // [H]
  const float* Wo     = (const float*)d_in[6];   // [H, V]
  const float* bo     = (const float*)d_in[7];   // [V]
  float* out = (float*)d_out;                    // logits [B,L,V] ++ final_hidden [B,H]

  // Workspace layout (floats): table | h0 | h1 | Whp (as v2f) | Wop (as v2f)
  float* table = (float*)d_ws;                              // V*H
  float* hbuf0 = table + (size_t)RNN_V * RNN_H;             // B*H
  float* hbuf1 = hbuf0 + (size_t)RNN_B * RNN_H;             // B*H
  v2f*   Whp   = (v2f*)(hbuf1 + (size_t)RNN_B * RNN_H);     // (H/2)*H v2f
  v2f*   Wop   = Whp + (size_t)(RNN_H / 2) * RNN_H;         // (H/2)*V v2f

  // One-time per call: pack weights into K-pair layout, build xin lookup table.
  pack_pairs_kernel<<<(RNN_H / 2) * RNN_H / 256, 256, 0, stream>>>(Wh, Whp, RNN_H, RNN_H);
  pack_pairs_kernel<<<(RNN_H / 2) * RNN_V / 256, 256, 0, stream>>>(Wo, Wop, RNN_H, RNN_V);
  embed_table_kernel<<<16, 256, 0, stream>>>(emb, We, bh, table);

  // s_0 = input hidden state.
  hipMemcpyAsync(hbuf0, hidden, (size_t)RNN_B * RNN_H * sizeof(float),
                 hipMemcpyDeviceToDevice, stream);

  // Serial recurrence: step t computes s_{t+1}; logits_{t-1} fused into step t.
  for (int t = 0; t < RNN_L; ++t) {
    const float* hi = (t & 1) ? hbuf1 : hbuf0;
    float*       ho = (t & 1) ? hbuf0 : hbuf1;
    rnn_step_kernel<<<17, 256, 0, stream>>>(hi, ho, table, x, Whp, Wop, bo, out, t);
  }
  // L is even -> s_L lives in hbuf0.
  const float* hfin = hbuf0;
  final_logits_kernel<<<1, 256, 0, stream>>>(hfin, Wop, bo, out);
  hipMemcpyAsync(out + (size_t)RNN_B * RNN_L * RNN_V, hfin,
                 (size_t)RNN_B * RNN_H * sizeof(float),
                 hipMemcpyDeviceToDevice, stream);
}